// MultiLossBce_68736656605337
// MI455X (gfx1250) — compile-verified
//
#include <hip/hip_runtime.h>
#include <math.h>

#define B_  16
#define C_  10
#define H_  400
#define W_  352
#define N_  50
#define HW_ (H_ * W_)     // 140800
#define HW4_ (HW_ / 4)    // 35200 float4 per plane
#define P_  64            // blocks per batch for the plane scan
#define BKG_TH_ 0.025f
#define LOG_CLAMP_ (-100.0f)

typedef float v8f __attribute__((ext_vector_type(8)));
typedef float v2f __attribute__((ext_vector_type(2)));

// log_sigmoid(x) = -softplus(-x) = -(max(-x,0) + log1p(exp(-|x|)))
__device__ __forceinline__ float log_sigmoid_(float x) {
    return -(fmaxf(-x, 0.0f) + log1pf(expf(-fabsf(x))));
}

// ---------------------------------------------------------------------------
// Kernel 1: per-(batch, stripe) partial reductions over channel-0 plane.
// partials[(b*P_+p)*4 + {0,1,2}] = { sum sigmoid(v), count(p>th), sum logv(sel) }
// ---------------------------------------------------------------------------
__global__ __launch_bounds__(256) void k1_bkg_partials(
    const float* __restrict__ out, float* __restrict__ partials) {
    const int b = blockIdx.x / P_;
    const int p = blockIdx.x % P_;
    const float4* plane = (const float4*)(out + (size_t)b * C_ * HW_);

    float sumP = 0.0f, cnt = 0.0f, sumLv = 0.0f;
    for (int i = p * 256 + threadIdx.x; i < HW4_; i += P_ * 256) {
        float4 q = plane[i];
        float vv[4] = {q.x, q.y, q.z, q.w};
#pragma unroll
        for (int k = 0; k < 4; ++k) {
            float v  = vv[k];
            float pr = 1.0f / (1.0f + expf(-v));           // sigmoid
            sumP += pr;
            // log_sigmoid(-v) clamped at -100
            float lv = fmaxf(-(fmaxf(v, 0.0f) + log1pf(expf(-fabsf(v)))), LOG_CLAMP_);
            bool sel = pr > BKG_TH_;                        // (almost always true)
            cnt   += sel ? 1.0f : 0.0f;
            sumLv += sel ? lv   : 0.0f;
        }
    }

    __shared__ float s0[256], s1[256], s2[256];
    const int tid = threadIdx.x;
    s0[tid] = sumP; s1[tid] = cnt; s2[tid] = sumLv;
    __syncthreads();
    for (int off = 128; off > 0; off >>= 1) {               // deterministic tree
        if (tid < off) {
            s0[tid] += s0[tid + off];
            s1[tid] += s1[tid + off];
            s2[tid] += s2[tid + off];
        }
        __syncthreads();
    }
    if (tid == 0) {
        float* dst = partials + (size_t)(b * P_ + p) * 4;
        dst[0] = s0[0]; dst[1] = s1[0]; dst[2] = s2[0]; dst[3] = 0.0f;
    }
}

// ---------------------------------------------------------------------------
// Kernel 2: one block per batch. Point gathers, dedup, per-batch loss terms.
// lossBuf[b*16 + 0..5] = six losses; 6..15 zero-padded for the WMMA finalize.
// ---------------------------------------------------------------------------
__global__ __launch_bounds__(64) void k2_per_batch(
    const float* __restrict__ out, const float* __restrict__ tgt,
    const float* __restrict__ partials, float* __restrict__ lossBuf) {
    const int b   = blockIdx.x;
    const int tid = threadIdx.x;
    const float* tb    = tgt + (size_t)b * N_ * 7;
    const size_t bBase = (size_t)b * C_ * HW_;

    __shared__ int   smLin[N_];
    __shared__ float sm[N_ * 9];

    if (tid < N_) {
        float tx = tb[tid * 7 + 0] * 2.0f;
        float ty = tb[tid * 7 + 1] * 2.0f;
        int dx = (int)floorf(tx);
        int dy = (int)floorf(ty);
        smLin[tid] = dx * W_ + dy;
    }
    __syncthreads();

    if (tid < N_) {
        const float x   = tb[tid * 7 + 0];
        const float y   = tb[tid * 7 + 1];
        const float z   = tb[tid * 7 + 2];
        const float dm0 = tb[tid * 7 + 3];
        const float dm1 = tb[tid * 7 + 4];
        const float dm2 = tb[tid * 7 + 5];
        const float ang = tb[tid * 7 + 6];

        const int lin = smLin[tid];
        bool first = true;
        for (int j = 0; j < tid; ++j)
            if (smLin[j] == lin) { first = false; break; }

        float g[10];
#pragma unroll
        for (int c = 0; c < 10; ++c) g[c] = out[bBase + (size_t)c * HW_ + lin];

        // --- subtraction terms for the masked background reductions ---
        float v0  = g[0];
        float p0  = 1.0f / (1.0f + expf(-v0));
        float lv0 = fmaxf(-(fmaxf(v0, 0.0f) + log1pf(expf(-fabsf(v0)))), LOG_CLAMP_);
        bool  s0  = p0 > BKG_TH_;
        float subP  = first ? p0 : 0.0f;
        float subC  = (first && s0) ? 1.0f : 0.0f;
        float subLv = (first && s0) ? lv0 : 0.0f;
        float firstF = first ? 1.0f : 0.0f;

        // --- objcls: -max(log_sigmoid(g0), -100)  (clsscale applied later) ---
        float obj = -fmaxf(log_sigmoid_(v0), LOG_CLAMP_);

        // --- xyz: (sigmoid(g[1:4]) - [fx,fy,tz])^2 ---
        float tx = x * 2.0f, ty = y * 2.0f;
        float t3[3] = {tx - floorf(tx), ty - floorf(ty), z * 0.25f};
        float xyz = 0.0f;
#pragma unroll
        for (int c = 0; c < 3; ++c) {
            float s = 1.0f / (1.0f + expf(-g[1 + c]));
            float d = s - t3[c];
            xyz += d * d;
        }

        // --- hwl: (g[4:7] - log(dims/mean))^2 ---
        const float mean[3] = {1.52f, 1.63f, 3.88f};
        const float dims[3] = {dm0, dm1, dm2};
        float hwl = 0.0f;
#pragma unroll
        for (int c = 0; c < 3; ++c) {
            float d = g[4 + c] - logf(dims[c] / mean[c]);
            hwl += d * d;
        }

        // --- angle class: -log_softmax(g[7:9])[anglecls] ---
        float a = g[7], bb = g[8];
        float mx  = fmaxf(a, bb);
        float lse = mx + logf(expf(a - mx) + expf(bb - mx));
        float acls = -((ang >= 0.0f) ? (bb - lse) : (a - lse));

        // --- angle sin: smooth-L1(sin(g9) - sin(sin(ang))) ---
        float dd = sinf(g[9]) - sinf(sinf(ang));
        float ad = fabsf(dd);
        float asn = (ad < 1.0f) ? 0.5f * dd * dd : ad - 0.5f;

        float* r = sm + tid * 9;
        r[0] = subP; r[1] = subC; r[2] = subLv; r[3] = firstF;
        r[4] = obj;  r[5] = xyz;  r[6] = hwl;   r[7] = acls;  r[8] = asn;
    }
    __syncthreads();

    if (tid == 0) {
        // deterministic sequential sums
        float sumP = 0.0f, cntA = 0.0f, sumLv = 0.0f;
        const float* pb = partials + (size_t)b * P_ * 4;
        for (int p = 0; p < P_; ++p) {
            sumP  += pb[p * 4 + 0];
            cntA  += pb[p * 4 + 1];
            sumLv += pb[p * 4 + 2];
        }
        float S[9] = {0, 0, 0, 0, 0, 0, 0, 0, 0};
        for (int n = 0; n < N_; ++n)
#pragma unroll
            for (int q = 0; q < 9; ++q) S[q] += sm[n * 9 + q];

        float D       = S[3];
        float nbkg    = (float)HW_ - D;
        float maskedP = sumP  - S[0];
        float cnt     = cntA  - S[1];
        float maskedL = sumLv - S[2];

        float clsscale = expf(-3.0f * maskedP / nbkg) * 2.5f;
        float bkg_l    = (cnt >= 0.5f) ? (-maskedL / fmaxf(cnt, 1.0f)) : 0.0f;
        float obj_l    = S[4] * (clsscale / (float)N_);
        float xyz_l    = S[5] / (3.0f * (float)N_);
        float hwl_l    = S[6] * (0.5f / (3.0f * (float)N_));
        float acls_l   = S[7] * (0.5f / (float)N_);
        float asin_l   = S[8] * (0.5f / (float)N_);

        float* dst = lossBuf + b * 16;
        dst[0] = bkg_l; dst[1] = obj_l; dst[2] = xyz_l;
        dst[3] = hwl_l; dst[4] = acls_l; dst[5] = asin_l;
        for (int i = 6; i < 16; ++i) dst[i] = 0.0f;
    }
}

// ---------------------------------------------------------------------------
// Kernel 3: one wave32. Batch-sum the 6 loss terms with exact fp32 WMMA:
// A[m][k] = loss[batch=4s+k][term=m] (rows 6..15 zero), B = ones =>
// D[m][n] = sum over batches of loss[b][m]. Four chained 16x16x4 WMMAs.
// ---------------------------------------------------------------------------
__global__ __launch_bounds__(32) void k3_finalize(
    const float* __restrict__ lossBuf, float* __restrict__ out7) {
    const int lane = threadIdx.x;          // 0..31, EXEC all ones
    const int m    = lane & 15;            // A-matrix row M
    const int klo  = (lane >> 4) << 1;     // lanes 0-15: K={0,1}; 16-31: K={2,3}

    v8f c = {};
    const v2f bones = {1.0f, 1.0f};
#pragma unroll
    for (int s = 0; s < 4; ++s) {          // batches 4s .. 4s+3
        v2f a;
        a.x = lossBuf[(4 * s + klo)     * 16 + m];   // A[m][klo]
        a.y = lossBuf[(4 * s + klo + 1) * 16 + m];   // A[m][klo+1]
        c = __builtin_amdgcn_wmma_f32_16x16x4_f32(
                false, a, false, bones, (short)0, c, false, false);
    }

    if (lane == 0) {                       // D[m][0] lives in c[m] of lane 0
        float total = c[0] + c[1] + c[2] + c[3] + c[4] + c[5];
        out7[0] = total;
        out7[1] = c[0]; out7[2] = c[1]; out7[3] = c[2];
        out7[4] = c[3]; out7[5] = c[4]; out7[6] = c[5];
    }
}

// ---------------------------------------------------------------------------
extern "C" void kernel_launch(void* const* d_in, const int* in_sizes, int n_in,
                              void* d_out, int out_size, void* d_ws, size_t ws_size,
                              hipStream_t stream) {
    const float* output = (const float*)d_in[0];   // (16,10,400,352) f32
    const float* target = (const float*)d_in[1];   // (16,50,7) f32
    float* ws       = (float*)d_ws;
    float* partials = ws;                          // B_*P_*4 = 4096 floats
    float* lossBuf  = ws + (size_t)B_ * P_ * 4;    // B_*16  =  256 floats

    k1_bkg_partials<<<dim3(B_ * P_), 256, 0, stream>>>(output, partials);
    k2_per_batch   <<<dim3(B_),       64, 0, stream>>>(output, target, partials, lossBuf);
    k3_finalize    <<<1,              32, 0, stream>>>(lossBuf, (float*)d_out);
}